// FFNO3d_48103633715618
// MI455X (gfx1250) — compile-verified
//
#include <hip/hip_runtime.h>
#include <math.h>

// ---------------------------------------------------------------------------
// FFNO-3D on MI455X (gfx1250): all matmuls via V_WMMA_F32_16X16X4_F32.
// wave32; A[16x4],B[4x16] = v2f (2 VGPR), C/D[16x16] = v8f (8 VGPR).
// Fragment layout per CDNA5 ISA 7.12.2:
//   A: lane%16 = M row; lanes<16 hold K={0,1}, lanes>=16 hold K={2,3}
//   B: lane%16 = N col; lanes<16 hold K={0,1}, lanes>=16 hold K={2,3}
//   C/D: VGPR r -> M = r + (lane<16 ? 0 : 8), N = lane%16
// ---------------------------------------------------------------------------

typedef __attribute__((ext_vector_type(2))) float v2f;
typedef __attribute__((ext_vector_type(8))) float v8f;

#define NB 4
#define NS 64
#define NW 32
#define NM 16
#define NPTS (NB * NS * NS * NS)          // 1,048,576 spatial points
#define HSZ ((size_t)NPTS * NW)           // floats per state tensor

__device__ __forceinline__ v8f wmma4(v2f a, v2f b, v8f c) {
  return __builtin_amdgcn_wmma_f32_16x16x4_f32(false, a, false, b, (short)0, c,
                                               false, false);
}

__device__ __forceinline__ v8f zero8() {
  v8f z;
  #pragma unroll
  for (int i = 0; i < 8; ++i) z[i] = 0.0f;
  return z;
}

__device__ __forceinline__ float gelu_exact(float x) {
  return 0.5f * x * (1.0f + erff(x * 0.70710678118654752440f));
}

// ---------------------------------------------------------------------------
// h[p][o] = sum_i x[p][i] * w[i][o] + b[o]    (K = C_IN = 4: one WMMA step)
// ---------------------------------------------------------------------------
__global__ __launch_bounds__(256) void k_input_proj(
    const float* __restrict__ x, const float* __restrict__ w,
    const float* __restrict__ bias, float* __restrict__ out) {
  const int lane = threadIdx.x & 31, wv = threadIdx.x >> 5;
  const int half = lane >> 4, l16 = lane & 15;
  const long pbase = ((long)blockIdx.x * 8 + wv) * 16;
  v2f a;
  a.x = x[(pbase + l16) * 4 + half * 2];
  a.y = x[(pbase + l16) * 4 + half * 2 + 1];
  v2f b0, b1;  // w layout [i(4)][o(32)]
  b0.x = w[(half * 2) * NW + l16];
  b0.y = w[(half * 2 + 1) * NW + l16];
  b1.x = w[(half * 2) * NW + 16 + l16];
  b1.y = w[(half * 2 + 1) * NW + 16 + l16];
  v8f d0 = zero8(), d1 = zero8();
  d0 = wmma4(a, b0, d0);
  d1 = wmma4(a, b1, d1);
  #pragma unroll
  for (int r = 0; r < 8; ++r) {
    long p = pbase + r + half * 8;
    out[p * NW + l16] = d0[r] + bias[l16];
    out[p * NW + 16 + l16] = d1[r] + bias[16 + l16];
  }
}

// ---------------------------------------------------------------------------
// Pointwise 32x32 matmul over all points.
//   WT=true : fc layout  w[i][o]   (h @ W)
//   WT=false: conv layout w[o][i]  (einsum 'bxyzi,oi->bxyzo')
// ---------------------------------------------------------------------------
template <bool WT, bool DOGELU>
__global__ __launch_bounds__(256) void k_pointwise(
    const float* __restrict__ hin, const float* __restrict__ w,
    const float* __restrict__ bias, float* __restrict__ out) {
  const int lane = threadIdx.x & 31, wv = threadIdx.x >> 5;
  const int half = lane >> 4, l16 = lane & 15;
  const long pbase = ((long)blockIdx.x * 8 + wv) * 16;
  v8f d0 = zero8(), d1 = zero8();
  #pragma unroll
  for (int kk = 0; kk < 8; ++kk) {
    const int i0 = kk * 4 + half * 2;
    v2f a;
    a.x = hin[(pbase + l16) * NW + i0];
    a.y = hin[(pbase + l16) * NW + i0 + 1];
    v2f b0, b1;
    if (WT) {
      b0.x = w[(i0)*NW + l16];      b0.y = w[(i0 + 1) * NW + l16];
      b1.x = w[(i0)*NW + 16 + l16]; b1.y = w[(i0 + 1) * NW + 16 + l16];
    } else {
      b0.x = w[(l16)*NW + i0];        b0.y = w[(l16)*NW + i0 + 1];
      b1.x = w[(16 + l16) * NW + i0]; b1.y = w[(16 + l16) * NW + i0 + 1];
    }
    d0 = wmma4(a, b0, d0);
    d1 = wmma4(a, b1, d1);
  }
  #pragma unroll
  for (int r = 0; r < 8; ++r) {
    long p = pbase + r + half * 8;
    float v0 = d0[r] + bias[l16];
    float v1 = d1[r] + bias[16 + l16];
    if (DOGELU) { v0 = gelu_exact(v0); v1 = gelu_exact(v1); }
    out[p * NW + l16] = v0;
    out[p * NW + 16 + l16] = v1;
  }
}

// ---------------------------------------------------------------------------
// Factorized spectral conv along one axis.
// One workgroup (8 waves) = 16 lines of 64 samples x 32 channels.
//   Stage1: G  = F(32x64)    * H(64x32)          (truncated ortho rDFT)
//   Stage2: G' = per-mode complex 32x32 mix, batched over 16 lines
//   Stage3: out+= Finv(64x32)* G'(32x32)         (truncated ortho irDFT)
// Dynamic LDS: F 8K | Finv 8K | G 64K | G' 64K = 144 KB (<= 320 KB/WGP).
// wgt layout: [in(32)][out(32)][mode(16)][2]  (one layer slice of fw*)
// ---------------------------------------------------------------------------
__global__ __launch_bounds__(256) void k_spectral_axis(
    const float* __restrict__ h, const float* __restrict__ wgt,
    float* __restrict__ acc, long strideU, long strideV, long strideA) {
  extern __shared__ float smem[];
  float* Ff = smem;                   // [32][64]
  float* Fi = smem + 2048;            // [64][32]
  float* G  = smem + 4096;            // [16][32][32]
  float* Gp = smem + 4096 + 16384;    // [16][32][32]

  const int tid = threadIdx.x;
  const int lane = tid & 31, wv = tid >> 5;
  const int half = lane >> 4, l16 = lane & 15;
  const float NRM = 0.125f;  // 1/sqrt(64)

  // Build DFT bases (normalization + Hermitian x2 folded in).
  for (int idx = tid; idx < 2048; idx += 256) {
    int m2 = idx >> 6, p = idx & 63, k = m2 >> 1;
    float ang = 6.28318530717958647693f * (float)(k * p) * (1.0f / 64.0f);
    Ff[idx] = (m2 & 1) ? (-__sinf(ang) * NRM) : (__cosf(ang) * NRM);
  }
  for (int idx = tid; idx < 2048; idx += 256) {
    int p = idx >> 5, m2 = idx & 31, k = m2 >> 1;
    float ang = 6.28318530717958647693f * (float)(k * p) * (1.0f / 64.0f);
    float sc = (k == 0 ? 1.0f : 2.0f) * NRM;
    Fi[idx] = (m2 & 1) ? (-__sinf(ang) * sc) : (__cosf(ang) * sc);
  }
  __syncthreads();

  // Each wave owns 2 of the workgroup's 16 lines.
  long base[2];
  #pragma unroll
  for (int ll = 0; ll < 2; ++ll) {
    int l = blockIdx.x * 16 + wv * 2 + ll;
    int b = l >> 12, u = (l >> 6) & 63, v = l & 63;
    base[ll] = (long)b * (64L * 64 * 64 * NW) + (long)u * strideU +
               (long)v * strideV;
  }

  // ---- Stage 1: forward truncated DFT ----
  #pragma unroll
  for (int ll = 0; ll < 2; ++ll) {
    const int li = wv * 2 + ll;
    v8f a00 = zero8(), a01 = zero8(), a10 = zero8(), a11 = zero8();
    for (int kk = 0; kk < 16; ++kk) {
      const int p0 = kk * 4 + half * 2;
      v2f b0, b1;
      b0.x = h[base[ll] + (long)p0 * strideA + l16];
      b0.y = h[base[ll] + (long)(p0 + 1) * strideA + l16];
      b1.x = h[base[ll] + (long)p0 * strideA + 16 + l16];
      b1.y = h[base[ll] + (long)(p0 + 1) * strideA + 16 + l16];
      v2f fa0, fa1;
      fa0.x = Ff[(l16)*64 + p0];      fa0.y = Ff[(l16)*64 + p0 + 1];
      fa1.x = Ff[(16 + l16) * 64 + p0]; fa1.y = Ff[(16 + l16) * 64 + p0 + 1];
      a00 = wmma4(fa0, b0, a00);
      a01 = wmma4(fa0, b1, a01);
      a10 = wmma4(fa1, b0, a10);
      a11 = wmma4(fa1, b1, a11);
    }
    float* g = G + li * 1024;
    #pragma unroll
    for (int r = 0; r < 8; ++r) {
      const int m = r + half * 8;
      g[(m)*32 + l16]           = a00[r];
      g[(m)*32 + 16 + l16]      = a01[r];
      g[(m + 16) * 32 + l16]    = a10[r];
      g[(m + 16) * 32 + 16 + l16] = a11[r];
    }
  }
  __syncthreads();

  // ---- Stage 2: per-mode complex channel mix (each wave: 2 modes) ----
  #pragma unroll
  for (int km = 0; km < 2; ++km) {
    const int k = wv + km * 8;
    v8f oRe0 = zero8(), oRe1 = zero8(), oIm0 = zero8(), oIm1 = zero8();
    for (int kk = 0; kk < 8; ++kk) {
      const int i0 = kk * 4 + half * 2, i1 = i0 + 1;
      v2f bRe, bIm;  // B[in][line]
      bRe.x = G[l16 * 1024 + (2 * k) * 32 + i0];
      bRe.y = G[l16 * 1024 + (2 * k) * 32 + i1];
      bIm.x = G[l16 * 1024 + (2 * k + 1) * 32 + i0];
      bIm.y = G[l16 * 1024 + (2 * k + 1) * 32 + i1];
      #pragma unroll
      for (int mt = 0; mt < 2; ++mt) {
        const int o = mt * 16 + l16;
        v2f aR, aI, aIn;  // A[out][in] = W^T, from [in][out][mode][2]
        aR.x = wgt[((i0 * NW + o) * NM + k) * 2 + 0];
        aR.y = wgt[((i1 * NW + o) * NM + k) * 2 + 0];
        aI.x = wgt[((i0 * NW + o) * NM + k) * 2 + 1];
        aI.y = wgt[((i1 * NW + o) * NM + k) * 2 + 1];
        aIn.x = -aI.x; aIn.y = -aI.y;  // f32 WMMA has no A-neg modifier
        if (mt == 0) {
          oRe0 = wmma4(aR, bRe, oRe0);  oRe0 = wmma4(aIn, bIm, oRe0);
          oIm0 = wmma4(aI, bRe, oIm0);  oIm0 = wmma4(aR, bIm, oIm0);
        } else {
          oRe1 = wmma4(aR, bRe, oRe1);  oRe1 = wmma4(aIn, bIm, oRe1);
          oIm1 = wmma4(aI, bRe, oIm1);  oIm1 = wmma4(aR, bIm, oIm1);
        }
      }
    }
    #pragma unroll
    for (int r = 0; r < 8; ++r) {
      const int o0 = r + half * 8;  // D: M = out-channel, N(lane) = line
      Gp[l16 * 1024 + (2 * k) * 32 + o0]          = oRe0[r];
      Gp[l16 * 1024 + (2 * k) * 32 + o0 + 16]     = oRe1[r];
      Gp[l16 * 1024 + (2 * k + 1) * 32 + o0]      = oIm0[r];
      Gp[l16 * 1024 + (2 * k + 1) * 32 + o0 + 16] = oIm1[r];
    }
  }
  __syncthreads();

  // ---- Stage 3: inverse truncated DFT, accumulate into acc ----
  #pragma unroll
  for (int ll = 0; ll < 2; ++ll) {
    const int li = wv * 2 + ll;
    const float* gp = Gp + li * 1024;
    v8f o00 = zero8(), o01 = zero8(), o10 = zero8(), o11 = zero8();
    v8f o20 = zero8(), o21 = zero8(), o30 = zero8(), o31 = zero8();
    for (int kk = 0; kk < 8; ++kk) {
      const int i0 = kk * 4 + half * 2;
      v2f b0, b1;
      b0.x = gp[(i0)*32 + l16];      b0.y = gp[(i0 + 1) * 32 + l16];
      b1.x = gp[(i0)*32 + 16 + l16]; b1.y = gp[(i0 + 1) * 32 + 16 + l16];
      v2f a0, a1, a2, a3;
      a0.x = Fi[(l16)*32 + i0];        a0.y = Fi[(l16)*32 + i0 + 1];
      a1.x = Fi[(16 + l16) * 32 + i0]; a1.y = Fi[(16 + l16) * 32 + i0 + 1];
      a2.x = Fi[(32 + l16) * 32 + i0]; a2.y = Fi[(32 + l16) * 32 + i0 + 1];
      a3.x = Fi[(48 + l16) * 32 + i0]; a3.y = Fi[(48 + l16) * 32 + i0 + 1];
      o00 = wmma4(a0, b0, o00);  o01 = wmma4(a0, b1, o01);
      o10 = wmma4(a1, b0, o10);  o11 = wmma4(a1, b1, o11);
      o20 = wmma4(a2, b0, o20);  o21 = wmma4(a2, b1, o21);
      o30 = wmma4(a3, b0, o30);  o31 = wmma4(a3, b1, o31);
    }
    #pragma unroll
    for (int r = 0; r < 8; ++r) {
      const int pr = r + half * 8;
      long i0a = base[ll] + (long)(pr)*strideA;
      long i1a = base[ll] + (long)(pr + 16) * strideA;
      long i2a = base[ll] + (long)(pr + 32) * strideA;
      long i3a = base[ll] + (long)(pr + 48) * strideA;
      acc[i0a + l16] += o00[r];  acc[i0a + 16 + l16] += o01[r];
      acc[i1a + l16] += o10[r];  acc[i1a + 16 + l16] += o11[r];
      acc[i2a + l16] += o20[r];  acc[i2a + 16 + l16] += o21[r];
      acc[i3a + l16] += o30[r];  acc[i3a + 16 + l16] += o31[r];
    }
  }
}

// ---------------------------------------------------------------------------
// Elementwise exact GELU (float4 vectorized).
// ---------------------------------------------------------------------------
__global__ __launch_bounds__(256) void k_gelu(const float* __restrict__ in,
                                              float* __restrict__ out) {
  const long i = (long)blockIdx.x * blockDim.x + threadIdx.x;
  float4 v = ((const float4*)in)[i];
  v.x = gelu_exact(v.x);
  v.y = gelu_exact(v.y);
  v.z = gelu_exact(v.z);
  v.w = gelu_exact(v.w);
  ((float4*)out)[i] = v;
}

// ---------------------------------------------------------------------------
// Final projection 32 -> 1.
// ---------------------------------------------------------------------------
__global__ __launch_bounds__(256) void k_fc3(const float* __restrict__ hin,
                                             const float* __restrict__ w,
                                             const float* __restrict__ bias,
                                             float* __restrict__ out) {
  const long p = (long)blockIdx.x * blockDim.x + threadIdx.x;
  const float4* hv = (const float4*)(hin + p * NW);
  const float4* wv = (const float4*)w;
  float s = 0.0f;
  #pragma unroll
  for (int i = 0; i < 8; ++i) {
    float4 a = hv[i], b = wv[i];
    s += a.x * b.x + a.y * b.y + a.z * b.z + a.w * b.w;
  }
  out[p] = s + bias[0];
}

// ---------------------------------------------------------------------------
extern "C" void kernel_launch(void* const* d_in, const int* in_sizes, int n_in,
                              void* d_out, int out_size, void* d_ws,
                              size_t ws_size, hipStream_t stream) {
  const float* x      = (const float*)d_in[0];
  const float* in_w   = (const float*)d_in[1];
  const float* in_b   = (const float*)d_in[2];
  const float* fwx    = (const float*)d_in[3];
  const float* fwy    = (const float*)d_in[4];
  const float* fwz    = (const float*)d_in[5];
  const float* conv_w = (const float*)d_in[6];
  const float* conv_b = (const float*)d_in[7];
  const float* fc1_w  = (const float*)d_in[8];
  const float* fc1_b  = (const float*)d_in[9];
  const float* fc2_w  = (const float*)d_in[10];
  const float* fc2_b  = (const float*)d_in[11];
  const float* fc3_w  = (const float*)d_in[12];
  const float* fc3_b  = (const float*)d_in[13];
  float* out = (float*)d_out;

  float* buf0 = (float*)d_ws;          // current h
  float* buf1 = buf0 + HSZ;            // layer accumulator

  const dim3 blk(256);
  const int wgPts = NPTS / (16 * 8);   // 8192 workgroups, 16 pts per wave
  const size_t smemB = (size_t)(4096 + 2 * 16384) * sizeof(float);  // 144 KB
  const long sX = 64L * 64 * NW;       // x stride (in floats)
  const long sY = 64L * NW;            // y stride
  const long sZ = (long)NW;            // z stride

  k_input_proj<<<wgPts, blk, 0, stream>>>(x, in_w, in_b, buf0);

  for (int i = 0; i < 4; ++i) {
    // acc = 1x1 conv (also initializes the accumulator)
    k_pointwise<false, false><<<wgPts, blk, 0, stream>>>(
        buf0, conv_w + i * (NW * NW), conv_b + i * NW, buf1);
    // spectral contributions per axis (reference order: z, y, x)
    const long wofs = (long)i * (NW * NW * NM * 2);
    k_spectral_axis<<<1024, blk, smemB, stream>>>(buf0, fwz + wofs, buf1,
                                                  sX, sY, sZ);
    k_spectral_axis<<<1024, blk, smemB, stream>>>(buf0, fwy + wofs, buf1,
                                                  sX, sZ, sY);
    k_spectral_axis<<<1024, blk, smemB, stream>>>(buf0, fwx + wofs, buf1,
                                                  sY, sZ, sX);
    if (i < 3)
      k_gelu<<<(int)(HSZ / 4 / 256), blk, 0, stream>>>(buf1, buf0);
  }

  // MLP head: buf1 holds last layer output (no gelu after layer 3)
  k_pointwise<true, true><<<wgPts, blk, 0, stream>>>(buf1, fc1_w, fc1_b, buf0);
  k_pointwise<true, true><<<wgPts, blk, 0, stream>>>(buf0, fc2_w, fc2_b, buf1);
  k_fc3<<<NPTS / 256, blk, 0, stream>>>(buf1, fc3_w, fc3_b, out);
}